// ESRNN_2568390443331
// MI455X (gfx1250) — compile-verified
//
#include <hip/hip_runtime.h>
#include <math.h>

typedef __attribute__((ext_vector_type(2))) float v2f;
typedef __attribute__((ext_vector_type(8))) float v8f;

// fp32 WMMA: D(16x16) = A(16x4) * B(4x16) + C   — wave32, CDNA5
#define WMMA_F32(A, B, C) \
  __builtin_amdgcn_wmma_f32_16x16x4_f32(false, (A), false, (B), (short)0, (C), false, false)

// Problem constants
constexpr int Bq   = 512;   // batch (series)
constexpr int Tq   = 300;   // time length
constexpr int Nw   = 289;   // T - I + 1 windows
constexpr int Nout = 277;   // T - I - O + 1
constexpr int TP   = 300;   // padded time rows in activation buffers (multiple of 12)

// ---------------------------------------------------------------------------
// Holt-Winters smoothing scan: one thread per series.
// levs: (B,300) row-major per series; seas: (B,312) row-major per series.
// ---------------------------------------------------------------------------
__global__ void hw_kernel(const float* __restrict__ train, const int* __restrict__ idxs,
                          const float* __restrict__ ilev, const float* __restrict__ iseas,
                          const float* __restrict__ iseason,
                          float* __restrict__ levs, float* __restrict__ seas)
{
  __shared__ float sb[256][12];
  int b = blockIdx.x * 256 + threadIdx.x;
  if (b >= Bq) return;
  int id = idxs[b];
  float ls = 1.f / (1.f + expf(-ilev[id]));
  float ss = 1.f / (1.f + expf(-iseas[id]));
  float* sl = sb[threadIdx.x];
  for (int s = 0; s < 12; s++) {
    float v = expf(iseason[id * 12 + s]);
    sl[s] = v;
    seas[b * 312 + s] = v;          // init_seas cols 0..11
  }
  float lev = train[b * 300] / sl[0];
  levs[b * 300] = lev;
  seas[b * 312 + 12] = sl[0];       // init_seas[:,0] at col 12
  for (int i = 1; i < 300; i++) {
    float y  = train[b * 300 + i];
    int   m  = i % 12;
    float si = sl[m];
    float nl = ls * (y / si) + (1.f - ls) * lev;
    float ns = ss * (y / nl) + (1.f - ss) * si;
    sl[m] = ns;
    lev   = nl;
    levs[b * 300 + i]      = nl;
    seas[b * 312 + 12 + i] = ns;    // cols 13..311
  }
}

// ---------------------------------------------------------------------------
// Build window_input X1 (TP,512,18) with zeroed pad rows [289,300),
// and window_output directly into d_out region 1.
// ---------------------------------------------------------------------------
__global__ void windows_kernel(const float* __restrict__ train, const float* __restrict__ info,
                               const float* __restrict__ levs, const float* __restrict__ seas,
                               float* __restrict__ X1, float* __restrict__ wout)
{
  int gid = blockIdx.x * blockDim.x + threadIdx.x;
  if (gid >= TP * Bq) return;
  int t = gid >> 9;
  int b = gid & 511;
  float* xp = X1 + ((size_t)t * Bq + b) * 18;
  if (t >= Nw) {                      // zero padding rows for dilation folds
    for (int i = 0; i < 18; i++) xp[i] = 0.f;
    return;
  }
  float lv = levs[b * 300 + t + 11];  // levs[:, I-1+n]
  for (int i = 0; i < 12; i++)
    xp[i] = (train[b * 300 + t + i] / seas[b * 312 + t + i]) / lv;
  for (int c = 0; c < 6; c++) xp[12 + c] = info[b * 6 + c];
  if (t < Nout) {
    float* wp = wout + ((size_t)t * Bq + b) * 12;
    for (int j = 0; j < 12; j++)
      wp[j] = (train[b * 300 + 12 + t + j] / seas[b * 312 + 12 + t + j]) / lv;
  }
}

// Zero rows [289,300) of a (300,512,64) activation buffer (fresh pad per layer).
__global__ void zero_pad_kernel(float* __restrict__ y)
{
  int gid = blockIdx.x * blockDim.x + threadIdx.x;
  const int total = (TP - Nw) * Bq * 64;
  if (gid < total) y[(size_t)Nw * Bq * 64 + gid] = 0.f;
}

// ---------------------------------------------------------------------------
// Dilated LSTM layer (dilation folded into Beff by the caller).
//   x viewed as (Tsteps, Beff, F1[+F2]), y as (Tsteps, Beff, 64).
//   Block = 16 batch rows, 8 waves; wave w computes gate cols [32w, 32w+32).
//   Weight B-fragments preloaded to registers, reused across all timesteps.
// ---------------------------------------------------------------------------
template <int F1, int F2>
__global__ __launch_bounds__(256) void lstm_kernel(
    const float* __restrict__ xA, const float* __restrict__ xB,
    const float* __restrict__ Wih, const float* __restrict__ Whh,
    const float* __restrict__ bih, const float* __restrict__ bhh,
    float* __restrict__ y, int Tsteps, int Beff)
{
  constexpr int F  = F1 + F2;
  constexpr int FP = (F + 3) & ~3;   // pad K to multiple of 4
  constexpr int KF = FP / 4;         // K-chunks over input features
  constexpr int KT = KF + 16;        // + 64/4 chunks over hidden

  __shared__ float x_s[16 * FP];
  __shared__ float h_s[16 * 64];
  __shared__ float g_s[16 * 256];

  const int tid  = threadIdx.x;
  const int lane = tid & 31;
  const int wave = tid >> 5;
  const int lm   = lane & 15;
  const int hi2  = (lane >= 16) ? 2 : 0;
  const int b0   = blockIdx.x * 16;
  const int n0   = (wave * 2)     * 16 + lm;   // gate column, tile 0
  const int n1   = (wave * 2 + 1) * 16 + lm;   // gate column, tile 1

  // Preload weight B-fragments (W^T, K-major 4-chunks) into registers.
  v2f wb0[KT], wb1[KT];
#pragma unroll
  for (int kk = 0; kk < KT; kk++) {
#pragma unroll
    for (int j = 0; j < 2; j++) {
      float v0, v1;
      if (kk < KF) {
        int k = kk * 4 + j + hi2;
        v0 = (k < F) ? Wih[n0 * F + k] : 0.f;
        v1 = (k < F) ? Wih[n1 * F + k] : 0.f;
      } else {
        int k = (kk - KF) * 4 + j + hi2;
        v0 = Whh[n0 * 64 + k];
        v1 = Whh[n1 * 64 + k];
      }
      wb0[kk][j] = v0;
      wb1[kk][j] = v1;
    }
  }

  // Per-thread persistent cell state: hidden col = tid&63, rows m = tid>>6 + 4q.
  const int hcol  = tid & 63;
  const int mrow0 = tid >> 6;
  float c_reg[4] = {0.f, 0.f, 0.f, 0.f};
  const float bi = bih[hcol]        + bhh[hcol];
  const float bf = bih[64  + hcol]  + bhh[64  + hcol];
  const float bg = bih[128 + hcol]  + bhh[128 + hcol];
  const float bo = bih[192 + hcol]  + bhh[192 + hcol];

  // init h = 0 and stage x[0]
  for (int e = tid; e < 16 * 64; e += 256) h_s[e] = 0.f;
  for (int e = tid; e < 16 * FP; e += 256) {
    int m = e / FP, f = e % FP;
    size_t row = (size_t)b0 + m;
    float v = 0.f;
    if (f < F1)     v = xA[row * F1 + f];
    else if (f < F) v = xB[row * F2 + (f - F1)];
    x_s[e] = v;
  }
  __syncthreads();

  for (int t = 0; t < Tsteps; t++) {
    // ---- gate GEMM via fp32 WMMA ----
    v8f acc0 = {0.f, 0.f, 0.f, 0.f, 0.f, 0.f, 0.f, 0.f};
    v8f acc1 = acc0;
#pragma unroll
    for (int kk = 0; kk < KT; kk++) {
      v2f a;
      if (kk < KF) {
        a[0] = x_s[lm * FP + kk * 4 + hi2];
        a[1] = x_s[lm * FP + kk * 4 + 1 + hi2];
      } else {
        int kb = (kk - KF) * 4;
        a[0] = h_s[lm * 64 + kb + hi2];
        a[1] = h_s[lm * 64 + kb + 1 + hi2];
      }
      acc0 = WMMA_F32(a, wb0[kk], acc0);
      acc1 = WMMA_F32(a, wb1[kk], acc1);
    }
    // accumulators -> LDS gate tile (C/D layout: vgpr r, lanes<16 -> M=r, else M=r+8)
#pragma unroll
    for (int r = 0; r < 8; r++) {
      int m = r + ((lane >= 16) ? 8 : 0);
      g_s[m * 256 + (wave * 2)     * 16 + lm] = acc0[r];
      g_s[m * 256 + (wave * 2 + 1) * 16 + lm] = acc1[r];
    }
    __syncthreads();

    // ---- elementwise LSTM cell (i,f,g,o gate order) ----
#pragma unroll
    for (int q = 0; q < 4; q++) {
      int m = mrow0 + 4 * q;
      float gi = g_s[m * 256 +        hcol] + bi;
      float gf = g_s[m * 256 + 64  +  hcol] + bf;
      float gg = g_s[m * 256 + 128 +  hcol] + bg;
      float go = g_s[m * 256 + 192 +  hcol] + bo;
      float iv = 1.f / (1.f + expf(-gi));
      float fv = 1.f / (1.f + expf(-gf));
      float gv = tanhf(gg);
      float ov = 1.f / (1.f + expf(-go));
      float c  = fv * c_reg[q] + iv * gv;
      float hv = ov * tanhf(c);
      c_reg[q] = c;
      h_s[m * 64 + hcol] = hv;
      y[((size_t)t * Beff + b0 + m) * 64 + hcol] = hv;
    }
    // stage next x while elementwise phase owns this barrier interval
    if (t + 1 < Tsteps) {
      for (int e = tid; e < 16 * FP; e += 256) {
        int m = e / FP, f = e % FP;
        size_t row = (size_t)(t + 1) * Beff + b0 + m;
        float v = 0.f;
        if (f < F1)     v = xA[row * F1 + f];
        else if (f < F) v = xB[row * F2 + (f - F1)];
        x_s[e] = v;
      }
    }
    __syncthreads();
  }
}

// ---------------------------------------------------------------------------
// Head: net = concat(out1,out2) (K=128) -> tanh(@nlW^T + nlb) -> @scW^T + scb.
// Writes net[:-12] to out0 and relu(net[288]*seas[:, -12:]*levs[:, -1]) to hold.
// grid.y = 278: rows 0..276 plus row 288.
// ---------------------------------------------------------------------------
__global__ __launch_bounds__(256) void head_kernel(
    const float* __restrict__ Y1, const float* __restrict__ Y2,
    const float* __restrict__ nlW, const float* __restrict__ nlb,
    const float* __restrict__ scW, const float* __restrict__ scb,
    const float* __restrict__ seas, const float* __restrict__ levs,
    float* __restrict__ out0, float* __restrict__ hold)
{
  __shared__ float a_s[16 * 128];
  __shared__ float hid_s[16 * 128];
  const int tid  = threadIdx.x;
  const int lane = tid & 31;
  const int wave = tid >> 5;
  const int lm   = lane & 15;
  const int hi2  = (lane >= 16) ? 2 : 0;
  const int b0   = blockIdx.x * 16;
  const int nrow = (blockIdx.y < Nout) ? blockIdx.y : (Nw - 1);

  for (int e = tid; e < 16 * 128; e += 256) {
    int m = e >> 7, f = e & 127;
    size_t row = (size_t)nrow * Bq + b0 + m;
    a_s[e] = (f < 64) ? Y1[row * 64 + f] : Y2[row * 64 + (f - 64)];
  }
  __syncthreads();

  { // stage 1: wave w -> hidden cols [16w, 16w+16)
    const int n = wave * 16 + lm;
    v8f acc = {0.f, 0.f, 0.f, 0.f, 0.f, 0.f, 0.f, 0.f};
#pragma unroll
    for (int kk = 0; kk < 32; kk++) {
      v2f a, bfr;
      a[0]   = a_s[lm * 128 + kk * 4 + hi2];
      a[1]   = a_s[lm * 128 + kk * 4 + 1 + hi2];
      bfr[0] = nlW[n * 128 + kk * 4 + hi2];
      bfr[1] = nlW[n * 128 + kk * 4 + 1 + hi2];
      acc = WMMA_F32(a, bfr, acc);
    }
    float bn = nlb[n];
#pragma unroll
    for (int r = 0; r < 8; r++) {
      int m = r + ((lane >= 16) ? 8 : 0);
      hid_s[m * 128 + n] = tanhf(acc[r] + bn);
    }
  }
  __syncthreads();

  if (wave == 0) { // stage 2: 12 output cols, K=128
    v8f acc = {0.f, 0.f, 0.f, 0.f, 0.f, 0.f, 0.f, 0.f};
    const bool colv = (lm < 12);
#pragma unroll
    for (int kk = 0; kk < 32; kk++) {
      v2f a, bfr;
      a[0]   = hid_s[lm * 128 + kk * 4 + hi2];
      a[1]   = hid_s[lm * 128 + kk * 4 + 1 + hi2];
      bfr[0] = colv ? scW[lm * 128 + kk * 4 + hi2] : 0.f;
      bfr[1] = colv ? scW[lm * 128 + kk * 4 + 1 + hi2] : 0.f;
      acc = WMMA_F32(a, bfr, acc);
    }
    if (colv) {
      float bn = scb[lm];
#pragma unroll
      for (int r = 0; r < 8; r++) {
        int m = r + ((lane >= 16) ? 8 : 0);
        int b = b0 + m;
        float v = acc[r] + bn;
        if (nrow < Nout)
          out0[((size_t)nrow * Bq + b) * 12 + lm] = v;
        if (nrow == Nw - 1) {
          float h = v * seas[b * 312 + 300 + lm] * levs[b * 300 + 299];
          hold[b * 12 + lm] = fmaxf(h, 0.f);
        }
      }
    }
  }
}

// ---------------------------------------------------------------------------
extern "C" void kernel_launch(void* const* d_in, const int* in_sizes, int n_in,
                              void* d_out, int out_size, void* d_ws, size_t ws_size,
                              hipStream_t stream)
{
  (void)in_sizes; (void)n_in; (void)out_size; (void)ws_size;
  const float* train   = (const float*)d_in[0];
  const float* val     = (const float*)d_in[1];
  // d_in[2] = test (unused by forward)
  const float* info    = (const float*)d_in[3];
  const int*   idxs    = (const int*)  d_in[4];
  const float* ilev    = (const float*)d_in[5];
  const float* iseas   = (const float*)d_in[6];
  const float* iseason = (const float*)d_in[7];
  const float* w1aWih  = (const float*)d_in[8];
  const float* w1aWhh  = (const float*)d_in[9];
  const float* w1abih  = (const float*)d_in[10];
  const float* w1abhh  = (const float*)d_in[11];
  const float* w1bWih  = (const float*)d_in[12];
  const float* w1bWhh  = (const float*)d_in[13];
  const float* w1bbih  = (const float*)d_in[14];
  const float* w1bbhh  = (const float*)d_in[15];
  const float* w2aWih  = (const float*)d_in[16];
  const float* w2aWhh  = (const float*)d_in[17];
  const float* w2abih  = (const float*)d_in[18];
  const float* w2abhh  = (const float*)d_in[19];
  const float* w2bWih  = (const float*)d_in[20];
  const float* w2bWhh  = (const float*)d_in[21];
  const float* w2bbih  = (const float*)d_in[22];
  const float* w2bbhh  = (const float*)d_in[23];
  const float* nlW     = (const float*)d_in[24];
  const float* nlb     = (const float*)d_in[25];
  const float* scW     = (const float*)d_in[26];
  const float* scb     = (const float*)d_in[27];

  float* ws   = (float*)d_ws;
  float* levs = ws;                               // 512*300
  float* seas = levs + 512 * 300;                 // 512*312
  float* X1   = seas + 512 * 312;                 // 300*512*18
  float* YA   = X1   + (size_t)300 * 512 * 18;    // 300*512*64  (Y1a, later reused as out2)
  float* Y1   = YA   + (size_t)300 * 512 * 64;    // 300*512*64  (out1)
  float* Y2A  = Y1   + (size_t)300 * 512 * 64;    // 300*512*64  (Y2a)

  float* out0  = (float*)d_out;                   // net[:-12]      277*512*12
  float* out1w = out0  + (size_t)Nout * 512 * 12; // window_output  277*512*12
  float* holdp = out1w + (size_t)Nout * 512 * 12; // hold_pred      512*12
  float* valO  = holdp + 512 * 12;                // val            512*12

  hw_kernel<<<2, 256, 0, stream>>>(train, idxs, ilev, iseas, iseason, levs, seas);
  windows_kernel<<<(TP * Bq + 255) / 256, 256, 0, stream>>>(train, info, levs, seas, X1, out1w);

  const int zgrid = ((TP - Nw) * Bq * 64 + 255) / 256;
  // layer 1a: dil=1  -> (289, 512, 18)
  lstm_kernel<18, 0><<<512 / 16, 256, 0, stream>>>(X1, nullptr, w1aWih, w1aWhh, w1abih, w1abhh, YA, 289, 512);
  zero_pad_kernel<<<zgrid, 256, 0, stream>>>(YA);
  // layer 1b: dil=3  -> (97, 1536, 64)
  lstm_kernel<64, 0><<<1536 / 16, 256, 0, stream>>>(YA, nullptr, w1bWih, w1bWhh, w1bbih, w1bbhh, Y1, 97, 1536);
  zero_pad_kernel<<<zgrid, 256, 0, stream>>>(Y1);
  // layer 2a: dil=6, skip-concat input (18+64) -> (49, 3072, 82)
  lstm_kernel<18, 64><<<3072 / 16, 256, 0, stream>>>(X1, Y1, w2aWih, w2aWhh, w2abih, w2abhh, Y2A, 49, 3072);
  zero_pad_kernel<<<zgrid, 256, 0, stream>>>(Y2A);
  // layer 2b: dil=12 -> (25, 6144, 64); output reuses YA
  lstm_kernel<64, 0><<<6144 / 16, 256, 0, stream>>>(Y2A, nullptr, w2bWih, w2bWhh, w2bbih, w2bbhh, YA, 25, 6144);

  head_kernel<<<dim3(32, Nout + 1), 256, 0, stream>>>(Y1, YA, nlW, nlb, scW, scb, seas, levs, out0, holdp);

  hipMemcpyAsync(valO, val, 512 * 12 * sizeof(float), hipMemcpyDeviceToDevice, stream);
}